// InterpolatingBSpline2d_8735963480089
// MI455X (gfx1250) — compile-verified
//
#include <hip/hip_runtime.h>
#include <stdint.h>

#define NC 4
#define MX 64
#define MY 64
#define NX (MX + 2)
#define NY (MY + 2)

typedef float v2f __attribute__((ext_vector_type(2)));
typedef float v8f __attribute__((ext_vector_type(8)));

// ---------------------------------------------------------------------------
// Pass 1: B-spline coefficient solve. The (M+2)x(M+2) natural-boundary system
// reduces analytically to: c[1]=d[0], c[M]=d[M-1], plus a constant-coefficient
// tridiagonal (1/6, 2/3, 1/6) Thomas solve for c[2..M-1], then
// c[0]=2c[1]-c[2], c[M+1]=2c[M]-c[M-1].  256 X-solves then 264 Y-solves;
// one block, __syncthreads between passes.
// ---------------------------------------------------------------------------
__device__ void thomas_solve(const float* __restrict__ d, int stride,
                             float* __restrict__ outp, int ostride, int M) {
  const float a = 1.0f / 6.0f;
  const float b = 2.0f / 3.0f;
  float cp[MX];
  float gp[MX];
  float c1 = d[0];
  float cM = d[(M - 1) * stride];
  // interior unknowns c[2..M-1]
  cp[2] = a / b;
  gp[2] = (d[1 * stride] - a * c1) / b;
#pragma unroll 1
  for (int i = 3; i <= M - 1; ++i) {
    float r = d[(i - 1) * stride];
    if (i == M - 1) r -= a * cM;
    float den = b - a * cp[i - 1];
    cp[i] = a / den;
    gp[i] = (r - a * gp[i - 1]) / den;
  }
  float sol[NX];
  sol[1] = c1;
  sol[M] = cM;
  sol[M - 1] = gp[M - 1];
#pragma unroll 1
  for (int i = M - 2; i >= 2; --i) sol[i] = gp[i] - cp[i] * sol[i + 1];
  sol[0] = 2.0f * sol[1] - sol[2];
  sol[M + 1] = 2.0f * sol[M] - sol[M - 1];
#pragma unroll 1
  for (int x = 0; x < M + 2; ++x) outp[x * ostride] = sol[x];
}

__global__ void coef_kernel(const float* __restrict__ data, float* __restrict__ ws) {
  float* coefs = ws;                      // NC*NX*NY floats
  float* cx = ws + NC * NX * NY;          // NC*NX*MY floats
  const int t = threadIdx.x;
  // X pass: one solve per (channel, y) column; reads stride MY
  if (t < NC * MY) {
    int c = t / MY, y = t % MY;
    thomas_solve(data + c * MX * MY + y, MY, cx + c * NX * MY + y, MY, MX);
  }
  __threadfence_block();
  __syncthreads();
  // Y pass: one solve per (channel, x') row; reads stride 1
  if (t < NC * NX) {
    int c = t / NX, x = t % NX;
    thomas_solve(cx + (c * NX + x) * MY, 1, coefs + (c * NX + x) * NY, 1, MY);
  }
}

// ---------------------------------------------------------------------------
// Cubic B-spline basis via WMMA: basis = BASIS(4x4) @ [t^3, t^2, t, 1].
// v_wmma_f32_16x16x4_f32: D(16x16) = A(16x4) x B(4x16) + C.
//   A rows M in {0..3} and {8..11} hold BASIS (replicated so both lane halves
//   of the D layout see rows 0..3 in VGPRs 0..3); other rows zero.  A is
//   loop-invariant: built ONCE per thread from pure ALU selects (no memory,
//   avoids the constant-buffer refetch the previous round showed in-loop).
//   B column n = powers of t for point n, built from per-lane t via shfl.
//   Two WMMAs per axis cover the wave's 32 points; each lane selects its half.
// ---------------------------------------------------------------------------
__device__ __forceinline__ float pick4(int r, float x0, float x1, float x2, float x3) {
  return r == 0 ? x0 : (r == 1 ? x1 : (r == 2 ? x2 : (r == 3 ? x3 : 0.0f)));
}

__device__ __forceinline__ v2f make_basis_A() {
  const int lane = threadIdx.x & 31;
  const int n = lane & 15;
  const bool lo = lane < 16;
  // BASIS row index for this lane's M position (rows 0..3 and 8..11 valid)
  int r = (n < 4) ? n : ((n >= 8 && n < 12) ? (n - 8) : -1);
  // BASIS columns (already /6): k0=(-1,3,-3,1)/6 k1=(3,-6,3,0)/6... transposed:
  //   BAS[r][0]: -1/6, 3/6, -3/6, 1/6
  //   BAS[r][1]:  3/6,-6/6,  3/6, 0
  //   BAS[r][2]: -3/6, 0/6,  3/6, 0
  //   BAS[r][3]:  1/6, 4/6,  1/6, 0
  float a0 = lo ? pick4(r, -1.0f / 6.0f, 3.0f / 6.0f, -3.0f / 6.0f, 1.0f / 6.0f)   // K=0
                : pick4(r, -3.0f / 6.0f, 0.0f, 3.0f / 6.0f, 0.0f);                 // K=2
  float a1 = lo ? pick4(r, 3.0f / 6.0f, -6.0f / 6.0f, 3.0f / 6.0f, 0.0f)           // K=1
                : pick4(r, 1.0f / 6.0f, 4.0f / 6.0f, 1.0f / 6.0f, 0.0f);           // K=3
  v2f A;
  A.x = a0;
  A.y = a1;
  return A;
}

__device__ __forceinline__ void basis_xy(float tx, float ty, v2f A,
                                         float bx[4], float by[4]) {
#if __has_builtin(__builtin_amdgcn_wmma_f32_16x16x4_f32)
  const int lane = threadIdx.x & 31;
  const int n = lane & 15;
  const bool lo = lane < 16;
  // Issue all four cross-lane reads up front so the dscnt waits batch.
  float tx0 = __shfl(tx, n, 32);
  float tx1 = __shfl(tx, n + 16, 32);
  float ty0 = __shfl(ty, n, 32);
  float ty1 = __shfl(ty, n + 16, 32);
  v2f BX0, BX1, BY0, BY1;
  BX0.x = lo ? tx0 * tx0 * tx0 : tx0 * tx0;  // K0: t^3 | K2: t^2
  BX0.y = lo ? tx0 : 1.0f;                   // K1: t   | K3: 1
  BX1.x = lo ? tx1 * tx1 * tx1 : tx1 * tx1;
  BX1.y = lo ? tx1 : 1.0f;
  BY0.x = lo ? ty0 * ty0 * ty0 : ty0 * ty0;
  BY0.y = lo ? ty0 : 1.0f;
  BY1.x = lo ? ty1 * ty1 * ty1 : ty1 * ty1;
  BY1.y = lo ? ty1 : 1.0f;
  v8f z = {0.f, 0.f, 0.f, 0.f, 0.f, 0.f, 0.f, 0.f};
  v8f dx0 = __builtin_amdgcn_wmma_f32_16x16x4_f32(false, A, false, BX0, (short)0, z, false, false);
  v8f dx1 = __builtin_amdgcn_wmma_f32_16x16x4_f32(false, A, false, BX1, (short)0, z, false, false);
  v8f dy0 = __builtin_amdgcn_wmma_f32_16x16x4_f32(false, A, false, BY0, (short)0, z, false, false);
  v8f dy1 = __builtin_amdgcn_wmma_f32_16x16x4_f32(false, A, false, BY1, (short)0, z, false, false);
  bx[0] = lo ? dx0[0] : dx1[0];
  bx[1] = lo ? dx0[1] : dx1[1];
  bx[2] = lo ? dx0[2] : dx1[2];
  bx[3] = lo ? dx0[3] : dx1[3];
  by[0] = lo ? dy0[0] : dy1[0];
  by[1] = lo ? dy0[1] : dy1[1];
  by[2] = lo ? dy0[2] : dy1[2];
  by[3] = lo ? dy0[3] : dy1[3];
#else
  float t2 = tx * tx, t3 = t2 * tx;
  bx[0] = (-t3 + 3.0f * t2 - 3.0f * tx + 1.0f) * (1.0f / 6.0f);
  bx[1] = (3.0f * t3 - 6.0f * t2 + 4.0f) * (1.0f / 6.0f);
  bx[2] = (-3.0f * t3 + 3.0f * t2 + 3.0f * tx + 1.0f) * (1.0f / 6.0f);
  bx[3] = t3 * (1.0f / 6.0f);
  float s2 = ty * ty, s3 = s2 * ty;
  by[0] = (-s3 + 3.0f * s2 - 3.0f * ty + 1.0f) * (1.0f / 6.0f);
  by[1] = (3.0f * s3 - 6.0f * s2 + 4.0f) * (1.0f / 6.0f);
  by[2] = (-s3 * 3.0f + 3.0f * s2 + 3.0f * ty + 1.0f) * (1.0f / 6.0f);
  by[3] = s3 * (1.0f / 6.0f);
#endif
}

__device__ __forceinline__ void axis_index(float un, int M, int& i, float& t) {
  i = (int)floorf(un);
  t = un - (float)i;
  if (un < 0.0f) {
    i = 0;
    t = un;
  }
  if (un >= (float)(M - 2)) {
    i = M - 2;
    t = un - (float)(M - 2);
  }
}

// ---------------------------------------------------------------------------
// Pass 2: evaluation. Coefs (69,696 B) staged global->LDS with async-to-LDS,
// then per-point 4x4x4ch gather + contraction from LDS.
// ---------------------------------------------------------------------------
__global__ void __launch_bounds__(256) interp_kernel(const float* __restrict__ u,
                                                     const float* __restrict__ coefs_g,
                                                     float* __restrict__ out, int npts) {
  __shared__ float sco[NC * NX * NY];  // 17424 floats = 69,696 B
  {
    const unsigned nvec = (NC * NX * NY) / 4;  // 4356 x 16B
    const unsigned ldsbase = (unsigned)(uintptr_t)(&sco[0]);  // LDS address = low 32 bits of the pointer
    const unsigned long long gbase = (unsigned long long)(uintptr_t)coefs_g;
    for (unsigned k = threadIdx.x; k < nvec; k += blockDim.x) {
      unsigned off = k * 16u;
      unsigned lptr = ldsbase + off;
      asm volatile("global_load_async_to_lds_b128 %0, %1, %2"
                   :: "v"(lptr), "v"(off), "s"(gbase)
                   : "memory");
    }
    asm volatile("s_wait_asynccnt 0" ::: "memory");
    __syncthreads();
  }

  const v2f A = make_basis_A();  // loop-invariant WMMA A operand, registers only

  const int gsz = gridDim.x * blockDim.x;
  const int gtid = blockIdx.x * blockDim.x + threadIdx.x;
  const int niter = (npts + gsz - 1) / gsz;  // uniform: keeps EXEC full for WMMA
  for (int it = 0; it < niter; ++it) {
    int idx = gtid + it * gsz;
    int idxc = idx < npts ? idx : npts - 1;  // clamp; only the store is predicated
    float2 uv = ((const float2*)u)[idxc];
    int ix, iy;
    float tx, ty;
    axis_index(uv.x * (float)(MX - 1), MX, ix, tx);
    axis_index(uv.y * (float)(MY - 1), MY, iy, ty);
    float bx[4], by[4];
    basis_xy(tx, ty, A, bx, by);
    float acc[NC];
#pragma unroll
    for (int c = 0; c < NC; ++c) {
      float s = 0.0f;
#pragma unroll
      for (int i = 0; i < 4; ++i) {
        const float* row = &sco[(c * NX + ix + i) * NY + iy];
        float v = fmaf(row[0], by[0],
                       fmaf(row[1], by[1], fmaf(row[2], by[2], row[3] * by[3])));
        s = fmaf(bx[i], v, s);
      }
      acc[c] = s;
    }
    if (idx < npts) {
      float4 o;
      o.x = acc[0];
      o.y = acc[1];
      o.z = acc[2];
      o.w = acc[3];
      ((float4*)out)[idx] = o;  // (B, C) row-contiguous, 16B/lane coalesced
    }
  }
}

extern "C" void kernel_launch(void* const* d_in, const int* in_sizes, int n_in,
                              void* d_out, int out_size, void* d_ws, size_t ws_size,
                              hipStream_t stream) {
  const float* u = (const float*)d_in[0];     // (B, 2)
  const float* data = (const float*)d_in[1];  // (4, 64, 64)
  float* out = (float*)d_out;                 // (B, 4)
  float* ws = (float*)d_ws;                   // [coefs: NC*NX*NY][cx: NC*NX*MY]
  const int npts = in_sizes[0] / 2;

  coef_kernel<<<1, 288, 0, stream>>>(data, ws);
  interp_kernel<<<2048, 256, 0, stream>>>(u, ws, out, npts);
}